// VGAEencoder_81870666596784
// MI455X (gfx1250) — compile-verified
//
#include <hip/hip_runtime.h>
#include <math.h>

typedef float v2f __attribute__((ext_vector_type(2)));
typedef float v8f __attribute__((ext_vector_type(8)));

#define NODES  20000
#define FDIM   3000
#define Z_DIM  128
#define H1DIM  512
#define G1DIM  32
#define OUTDIM 3
#define EDGES  640000
#define PE_ALPHA 0.1f

// ---------------------------------------------------------------------------
// fp32 GEMM via V_WMMA_F32_16X16X4_F32. One wave computes a 32x32 output
// tile: 2 A fragments x 2 B fragments -> 4 accumulators, so each K-step
// issues 2 A(b64) + 4 B(b32) loads for 4 WMMAs. M%32==0, N%32==0, K%4==0
// hold at every call site; the tile guard is wave-uniform so EXEC is
// all-ones at every WMMA (ISA 7.12 requirement).
// Fragment layouts (CDNA5 ISA):
//   A 16x4  : lane -> row lane%16, VGPR v -> K = (lane/16)*2 + v
//   B 4x16  : lane -> col lane%16, VGPR v -> K = (lane/16)*2 + v
//   C/D     : lane l, VGPR v -> row v + (l/16)*8, col l%16
// ---------------------------------------------------------------------------
__global__ void gemm_wmma_f32(const float* __restrict__ A,
                              const float* __restrict__ B,
                              const float* __restrict__ bias,
                              float* __restrict__ C,
                              int M, int N, int K, int relu)
{
    const int lane = threadIdx.x & 31;
    const int wib  = threadIdx.x >> 5;
    const int gw   = blockIdx.x * (blockDim.x >> 5) + wib;
    const int nt2  = N >> 5;                 // 32-wide column groups
    const int mt   = gw / nt2;
    const int nt   = gw - mt * nt2;
    if (mt >= (M >> 5)) return;              // wave-uniform

    const int row0 = mt << 5;                // 32 rows
    const int col0 = nt << 5;                // 32 cols
    const int kb   = (lane >> 4) << 1;       // K sub-offset per lane half
    const int n0   = col0 + (lane & 15);
    const int n1   = n0 + 16;

    const float* __restrict__ Arow0 = A + (size_t)(row0 + (lane & 15)) * K;
    const float* __restrict__ Arow1 = Arow0 + (size_t)16 * K;

    v8f acc00 = {0.f,0.f,0.f,0.f,0.f,0.f,0.f,0.f};
    v8f acc01 = {0.f,0.f,0.f,0.f,0.f,0.f,0.f,0.f};
    v8f acc10 = {0.f,0.f,0.f,0.f,0.f,0.f,0.f,0.f};
    v8f acc11 = {0.f,0.f,0.f,0.f,0.f,0.f,0.f,0.f};

    int bidx = kb * N + n0;                  // 32-bit B index (K*N < 2^31)
    for (int k0 = 0; k0 < K; k0 += 4) {
        v2f a0, a1, b0, b1;
        a0.x = Arow0[k0 + kb];  a0.y = Arow0[k0 + kb + 1];
        a1.x = Arow1[k0 + kb];  a1.y = Arow1[k0 + kb + 1];
        b0.x = B[bidx];         b0.y = B[bidx + N];
        b1.x = B[bidx + 16];    b1.y = B[bidx + N + 16];
        acc00 = __builtin_amdgcn_wmma_f32_16x16x4_f32(
                    false, a0, false, b0, (short)0, acc00, false, false);
        acc01 = __builtin_amdgcn_wmma_f32_16x16x4_f32(
                    false, a0, false, b1, (short)0, acc01, false, false);
        acc10 = __builtin_amdgcn_wmma_f32_16x16x4_f32(
                    false, a1, false, b0, (short)0, acc10, false, false);
        acc11 = __builtin_amdgcn_wmma_f32_16x16x4_f32(
                    false, a1, false, b1, (short)0, acc11, false, false);
        bidx += 4 * N;
    }

    const float bv0 = bias ? bias[n0] : 0.0f;
    const float bv1 = bias ? bias[n1] : 0.0f;
    const int rbase0 = row0 + ((lane >> 4) << 3);
    const int rbase1 = rbase0 + 16;
#pragma unroll
    for (int v = 0; v < 8; ++v) {
        float x00 = acc00[v] + bv0;
        float x01 = acc01[v] + bv1;
        float x10 = acc10[v] + bv0;
        float x11 = acc11[v] + bv1;
        if (relu) {
            x00 = fmaxf(x00, 0.0f); x01 = fmaxf(x01, 0.0f);
            x10 = fmaxf(x10, 0.0f); x11 = fmaxf(x11, 0.0f);
        }
        C[(size_t)(rbase0 + v) * N + n0] = x00;
        C[(size_t)(rbase0 + v) * N + n1] = x01;
        C[(size_t)(rbase1 + v) * N + n0] = x10;
        C[(size_t)(rbase1 + v) * N + n1] = x11;
    }
}

// po_emb = fe + 0.1 * geom_lowf Fourier PE of 2D coords (zdim=128)
__global__ void pe_add_kernel(const float* __restrict__ fe,
                              const float* __restrict__ coords,
                              float* __restrict__ po)
{
    const int tid = blockIdx.x * blockDim.x + threadIdx.x;
    if (tid >= NODES * Z_DIM) return;
    const int n = tid >> 7;          // /128
    const int c = tid & 127;
    const int a = c >> 6;            // axis 0/1
    const int j = c & 63;            // 0..31 sin, 32..63 cos
    const int i = j & 31;
    // freqs[i] = 64 * (1/64)^(i/31) = 64 * 2^(-6 i / 31)
    const float freq = 64.0f * exp2f(-6.0f * (float)i / 31.0f);
    const float k = coords[n * 2 + a] * freq;
    const float p = (j < 32) ? sinf(k) : cosf(k);
    po[tid] = fe[tid] + PE_ALPHA * p;
}

__global__ void deg_init_kernel(float* __restrict__ deg)
{
    const int n = blockIdx.x * blockDim.x + threadIdx.x;
    if (n < NODES) deg[n] = 1.0f;            // self-loop contribution
}

__global__ void deg_acc_kernel(const int* __restrict__ dst, float* __restrict__ deg)
{
    const int e = blockIdx.x * blockDim.x + threadIdx.x;
    if (e < EDGES) atomicAdd(&deg[dst[e]], 1.0f);
}

__global__ void deg_rsqrt_kernel(float* __restrict__ deg)
{
    const int n = blockIdx.x * blockDim.x + threadIdx.x;
    if (n < NODES) deg[n] = rsqrtf(deg[n]);  // deg >= 1 always
}

__global__ void zero_kernel(float* __restrict__ p, int n)
{
    const int i = blockIdx.x * blockDim.x + threadIdx.x;
    if (i < n) p[i] = 0.0f;
}

// one wave per edge: 32 coalesced channels
__global__ void scatter32_kernel(const int* __restrict__ src,
                                 const int* __restrict__ dst,
                                 const float* __restrict__ dinv,
                                 const float* __restrict__ h,
                                 float* __restrict__ agg)
{
    const long tid = (long)blockIdx.x * blockDim.x + threadIdx.x;
    if (tid >= (long)EDGES * G1DIM) return;
    const int e = (int)(tid >> 5);
    const int c = (int)(tid & 31);
    const int s = src[e], d = dst[e];
    const float w = dinv[s] * dinv[d];
    atomicAdd(&agg[(size_t)d * G1DIM + c], h[(size_t)s * G1DIM + c] * w);
}

// h1 = tanh(agg + h * dinv^2)  (adds the self-loop term, in-place on agg)
__global__ void tanh_self_kernel(const float* __restrict__ h,
                                 const float* __restrict__ dinv,
                                 float* __restrict__ agg)
{
    const int tid = blockIdx.x * blockDim.x + threadIdx.x;
    if (tid >= NODES * G1DIM) return;
    const int n = tid >> 5;
    const float di = dinv[n];
    agg[tid] = tanhf(agg[tid] + h[tid] * di * di);
}

// mu_pre = h1 @ W_gc2 ; ls_pre = h1 @ W_gc3   (32 -> 3, per node)
__global__ void head_mm_kernel(const float* __restrict__ h1,
                               const float* __restrict__ W2,
                               const float* __restrict__ W3,
                               float* __restrict__ mup,
                               float* __restrict__ lsp)
{
    const int n = blockIdx.x * blockDim.x + threadIdx.x;
    if (n >= NODES) return;
    const float* __restrict__ r = h1 + (size_t)n * G1DIM;
    float a2[3] = {0.f, 0.f, 0.f};
    float a3[3] = {0.f, 0.f, 0.f};
#pragma unroll
    for (int k = 0; k < G1DIM; ++k) {
        const float hv = r[k];
        a2[0] += hv * W2[k * 3 + 0];
        a2[1] += hv * W2[k * 3 + 1];
        a2[2] += hv * W2[k * 3 + 2];
        a3[0] += hv * W3[k * 3 + 0];
        a3[1] += hv * W3[k * 3 + 1];
        a3[2] += hv * W3[k * 3 + 2];
    }
#pragma unroll
    for (int c = 0; c < 3; ++c) {
        mup[(size_t)n * 3 + c] = a2[c];
        lsp[(size_t)n * 3 + c] = a3[c];
    }
}

// initialize outputs with the self-loop term
__global__ void out_init_kernel(const float* __restrict__ mup,
                                const float* __restrict__ lsp,
                                const float* __restrict__ dinv,
                                float* __restrict__ mu_out,
                                float* __restrict__ ls_out)
{
    const int tid = blockIdx.x * blockDim.x + threadIdx.x;
    if (tid >= NODES * OUTDIM) return;
    const int n = tid / 3;
    const float w = dinv[n] * dinv[n];
    mu_out[tid] = mup[tid] * w;
    ls_out[tid] = lsp[tid] * w;
}

// 6 atomics per edge (3 mu + 3 logstd channels)
__global__ void scatter3_kernel(const int* __restrict__ src,
                                const int* __restrict__ dst,
                                const float* __restrict__ dinv,
                                const float* __restrict__ mup,
                                const float* __restrict__ lsp,
                                float* __restrict__ mu_out,
                                float* __restrict__ ls_out)
{
    const long tid = (long)blockIdx.x * blockDim.x + threadIdx.x;
    if (tid >= (long)EDGES * 6) return;
    const int e = (int)(tid / 6);
    const int j = (int)(tid - (long)e * 6);
    const int s = src[e], d = dst[e];
    const float w = dinv[s] * dinv[d];
    if (j < 3) atomicAdd(&mu_out[(size_t)d * 3 + j],       mup[(size_t)s * 3 + j] * w);
    else       atomicAdd(&ls_out[(size_t)d * 3 + (j - 3)], lsp[(size_t)s * 3 + (j - 3)] * w);
}

extern "C" void kernel_launch(void* const* d_in, const int* in_sizes, int n_in,
                              void* d_out, int out_size, void* d_ws, size_t ws_size,
                              hipStream_t stream) {
    const float* x      = (const float*)d_in[0];
    const float* coords = (const float*)d_in[1];
    const float* W_fen1 = (const float*)d_in[2];
    const float* b_fen1 = (const float*)d_in[3];
    const float* W_fen2 = (const float*)d_in[4];
    const float* b_fen2 = (const float*)d_in[5];
    const float* W_gc1  = (const float*)d_in[6];
    const float* W_gc2  = (const float*)d_in[7];
    const float* W_gc3  = (const float*)d_in[8];
    const int*   ei     = (const int*)d_in[9];
    const int* src = ei;
    const int* dst = ei + EDGES;

    float* out    = (float*)d_out;
    float* mu_out = out;                      // 60000
    float* ls_out = out + NODES * OUTDIM;     // 60000
    float* po_out = out + 2 * NODES * OUTDIM; // 2,560,000

    // workspace layout (floats). fe1 region (10.24M) is dead after GEMM2 and
    // is reused for every GCN-stage buffer.
    float* ws   = (float*)d_ws;
    float* fe1  = ws;                         // 20000*512
    float* fe   = ws + 10240000;              // 20000*128
    float* deg  = ws;                         // 20000        (reuse fe1)
    float* h    = ws + 20480;                 // 20000*32
    float* agg  = ws + 660480;                // 20000*32
    float* mup  = ws + 1300480;               // 20000*3
    float* lsp  = ws + 1360480;               // 20000*3

    const int TB = 256;

    // GEMM1: fe1 = relu(x @ W_fen1 + b_fen1)   M=20000 N=512 K=3000
    {
        const int waves = (NODES / 32) * (H1DIM / 32); // 10000
        gemm_wmma_f32<<<(waves + 3) / 4, 128, 0, stream>>>(
            x, W_fen1, b_fen1, fe1, NODES, H1DIM, FDIM, 1);
    }
    // GEMM2: fe = relu(fe1 @ W_fen2 + b_fen2)  M=20000 N=128 K=512
    {
        const int waves = (NODES / 32) * (Z_DIM / 32); // 2500
        gemm_wmma_f32<<<(waves + 3) / 4, 128, 0, stream>>>(
            fe1, W_fen2, b_fen2, fe, NODES, Z_DIM, H1DIM, 1);
    }
    // po_emb = fe + 0.1 * PE
    pe_add_kernel<<<(NODES * Z_DIM + TB - 1) / TB, TB, 0, stream>>>(fe, coords, po_out);

    // degrees -> dinv  (fe1 region is free now)
    deg_init_kernel<<<(NODES + TB - 1) / TB, TB, 0, stream>>>(deg);
    deg_acc_kernel<<<(EDGES + TB - 1) / TB, TB, 0, stream>>>(dst, deg);
    deg_rsqrt_kernel<<<(NODES + TB - 1) / TB, TB, 0, stream>>>(deg);

    // GEMM3: h = fe @ W_gc1   M=20000 N=32 K=128
    {
        const int waves = (NODES / 32) * (G1DIM / 32); // 625
        gemm_wmma_f32<<<(waves + 3) / 4, 128, 0, stream>>>(
            fe, W_gc1, (const float*)nullptr, h, NODES, G1DIM, Z_DIM, 0);
    }

    // layer-1 aggregation: agg = sum_{e} norm * h[src], then h1 = tanh(agg + self)
    zero_kernel<<<(NODES * G1DIM + TB - 1) / TB, TB, 0, stream>>>(agg, NODES * G1DIM);
    {
        const long total = (long)EDGES * G1DIM;
        scatter32_kernel<<<(int)((total + TB - 1) / TB), TB, 0, stream>>>(src, dst, deg, h, agg);
    }
    tanh_self_kernel<<<(NODES * G1DIM + TB - 1) / TB, TB, 0, stream>>>(h, deg, agg);

    // heads: mu_pre / logstd_pre
    head_mm_kernel<<<(NODES + TB - 1) / TB, TB, 0, stream>>>(agg, W_gc2, W_gc3, mup, lsp);

    // layer-2 aggregation into outputs
    out_init_kernel<<<(NODES * OUTDIM + TB - 1) / TB, TB, 0, stream>>>(mup, lsp, deg, mu_out, ls_out);
    {
        const long total = (long)EDGES * 6;
        scatter3_kernel<<<(int)((total + TB - 1) / TB), TB, 0, stream>>>(src, dst, deg, mup, lsp, mu_out, ls_out);
    }
}